// QuotientWisard_67147518705987
// MI455X (gfx1250) — compile-verified
//
#include <hip/hip_runtime.h>

typedef __attribute__((ext_vector_type(2))) float v2f;
typedef __attribute__((ext_vector_type(8))) float v8f;

#define N_NEURONS   1024
#define TABLE_SIZE  65536
#define BATCH       8192
#define SPB         16      // samples per block

// Block: 256 threads = 8 waves. Block b handles samples [16b, 16b+16).
// Wave w handles neurons [128w, 128w+128); lane l -> sample (l&15),
// neurons jbase..jbase+63 with jbase = 128w + 64*(l>>4).
__global__ __launch_bounds__(256) void qwisard_gather_wmma(
    const int* __restrict__ data,      // [BATCH, N_NEURONS] int32
    const float* __restrict__ table,   // [N_NEURONS, TABLE_SIZE] f32
    float* __restrict__ out)           // [BATCH] f32
{
    __shared__ float s_part[8 * SPB];  // [wave][sample]

    const int tid  = threadIdx.x;
    const int wave = tid >> 5;
    const int lane = tid & 31;
    const int s0   = blockIdx.x * SPB;

    const int samp  = lane & 15;
    const int half  = lane >> 4;
    const int jbase = wave * 128 + half * 64;

    // Contiguous 256B index run for this lane (16B-aligned: jbase % 64 == 0).
    const int4* idx4 = (const int4*)(data + (size_t)(s0 + samp) * N_NEURONS + jbase);
    // Table rows for this lane start at neuron jbase; within the run,
    // element offset = ((t) << 16) | index  (TABLE_SIZE == 1<<16).
    const float* pbase = table + (size_t)jbase * TABLE_SIZE;

    float acc0 = 0.f, acc1 = 0.f, acc2 = 0.f, acc3 = 0.f;
#pragma unroll
    for (int t = 0; t < 64; t += 4) {
        const int4 id = idx4[t >> 2];
        acc0 += pbase[(size_t)(((unsigned)(t + 0) << 16) | (unsigned)id.x)];
        acc1 += pbase[(size_t)(((unsigned)(t + 1) << 16) | (unsigned)id.y)];
        acc2 += pbase[(size_t)(((unsigned)(t + 2) << 16) | (unsigned)id.z)];
        acc3 += pbase[(size_t)(((unsigned)(t + 3) << 16) | (unsigned)id.w)];
    }
    const float partial = (acc0 + acc1) + (acc2 + acc3);

    // Wave-wide reduction via V_WMMA_F32_16X16X4_F32.
    // A (16x4): VGPR0 = {K0 (lanes 0-15), K2 (lanes 16-31)}, VGPR1 = {K1, K3}.
    // With a = (partial, 0): A[M,0] = partial[lane M], A[M,2] = partial[lane M+16].
    // B = all ones -> D[M,N] = partial[M] + partial[M+16] for every column N,
    // i.e. the full per-sample sum over this wave's 128 neurons.
    v2f a; a.x = partial; a.y = 0.0f;
    v2f b; b.x = 1.0f;    b.y = 1.0f;
    v8f c = {};
    v8f d = __builtin_amdgcn_wmma_f32_16x16x4_f32(
        /*neg_a=*/false, a, /*neg_b=*/false, b,
        /*c_mod=*/(short)0, c, /*reuse_a=*/false, /*reuse_b=*/false);

    // C/D layout: VGPR r holds row M=r (lanes 0-15) and M=r+8 (lanes 16-31).
    if (lane == 0) {
#pragma unroll
        for (int r = 0; r < 8; ++r) s_part[wave * SPB + r] = d[r];
    } else if (lane == 16) {
#pragma unroll
        for (int r = 0; r < 8; ++r) s_part[wave * SPB + 8 + r] = d[r];
    }
    __syncthreads();

    // Sum the 8 wave-partials per sample and store.
    if (tid < SPB) {
        float r = 0.f;
#pragma unroll
        for (int w = 0; w < 8; ++w) r += s_part[w * SPB + tid];
        out[s0 + tid] = r;
    }
}

extern "C" void kernel_launch(void* const* d_in, const int* in_sizes, int n_in,
                              void* d_out, int out_size, void* d_ws, size_t ws_size,
                              hipStream_t stream) {
    (void)in_sizes; (void)n_in; (void)d_ws; (void)ws_size; (void)out_size;
    const int*   data  = (const int*)d_in[0];
    const float* table = (const float*)d_in[1];
    float*       out   = (float*)d_out;

    qwisard_gather_wmma<<<dim3(BATCH / SPB), dim3(256), 0, stream>>>(data, table, out);
}